// Cnn_map_1752346656962
// MI455X (gfx1250) — compile-verified
//
#include <hip/hip_runtime.h>
#include <hip/hip_bf16.h>

typedef float v2f __attribute__((ext_vector_type(2)));
typedef float v8f __attribute__((ext_vector_type(8)));

#define Bn 8
#define Cn 64
#define Hn 256
#define Wn 512
#define TAPS 9
#define CHW ((long)Cn * Hn * Wn)
#define HW  ((long)Hn * Wn)

// One block per (b, h) output row (pre-upsample). 256 threads = 8 wave32.
// Each wave handles 4 groups of 16 pixels; channel reduction runs through
// v_wmma_f32_16x16x4_f32 (fp32 in / fp32 accum => bit-compatible numerics).
__global__ __launch_bounds__(256)
void sphere_conv_wmma_kernel(const float* __restrict__ feat,
                             const float* __restrict__ wgt,   // [C,3,3] flat: c*9+t
                             const int*   __restrict__ gi,    // [H,W,9]
                             const int*   __restrict__ gj,    // [H,W,9]
                             float*       __restrict__ out)   // [B,1,2H,2W]
{
    __shared__ float w_lds[Cn * TAPS];   // 576 floats
    __shared__ int   gi_lds[TAPS];

    const int tid = threadIdx.x;
    const int bh  = blockIdx.x;          // 0 .. B*H-1
    const int b   = bh >> 8;             // H = 256
    const int h   = bh & (Hn - 1);

    // Stage weights + per-row tap row-indices (gi is constant along w).
    for (int i = tid; i < Cn * TAPS; i += 256) w_lds[i] = wgt[i];
    if (tid < TAPS) gi_lds[tid] = gi[(long)h * Wn * TAPS + tid];   // gi[h,0,t]
    __syncthreads();

    const int wave = tid >> 5;           // 0..7
    const int lane = tid & 31;
    const int half = lane >> 4;          // 0: K-slots {0,1}, 1: K-slots {2,3}
    const int lp   = lane & 15;          // pixel-in-group == M/N index

    const long fb = (long)b * CHW;       // batch base into feature

    for (int g = 0; g < 4; ++g) {
        const int p = (wave * 4 + g) * 16 + lp;          // pixel 0..511

        // Gather columns for this pixel, all 9 taps (lane-adjacent => coalesced).
        int col[TAPS];
        const long gbase = ((long)h * Wn + p) * TAPS;
        #pragma unroll
        for (int t = 0; t < TAPS; ++t) col[t] = gj[gbase + t];

        v8f acc = {};                                    // C/D accumulator (f32)

        #pragma unroll 1
        for (int t = 0; t < TAPS; ++t) {
            const long rowbase = fb + (long)gi_lds[t] * Wn + col[t];
            const float* wt = &w_lds[t];                 // wt[c*9] = weight[c, t]

            #pragma unroll
            for (int cc = 0; cc < 16; ++cc) {
                // Physical K-slot convention (identical for A and B):
                //   slot = 2*laneHalf + vgprIndex  ->  channel 4*cc + slot
                const int c0 = cc * 4 + half * 2;

                v2f a, bm;
                a.x  = wt[(c0 + 0) * TAPS];              // A: weight, replicated over M
                a.y  = wt[(c0 + 1) * TAPS];
                bm.x = feat[rowbase + (long)(c0 + 0) * HW];  // B: gathered data, N=pixel
                bm.y = feat[rowbase + (long)(c0 + 1) * HW];

                acc = __builtin_amdgcn_wmma_f32_16x16x4_f32(
                        /*neg_a=*/false, a, /*neg_b=*/false, bm,
                        /*c_mod=*/(short)0, acc,
                        /*reuse_a=*/false, /*reuse_b=*/false);
            }
        }

        // Every M-row of D is identical; D vgpr0: lanes0-15 -> (M=0,N=lane),
        // lanes16-31 -> (M=8,N=lane-16). So acc[0] in each lane is the result
        // for pixel (lane & 15).
        const float v = acc[0];

        // 2x2 upsample: laneHalf selects dy, float2 store covers dx=0,1.
        const long orow = (long)b * (2 * Hn) + 2 * h + half;
        float2 o2; o2.x = v; o2.y = v;
        *(float2*)&out[orow * (2 * Wn) + 2 * p] = o2;
    }
}

extern "C" void kernel_launch(void* const* d_in, const int* in_sizes, int n_in,
                              void* d_out, int out_size, void* d_ws, size_t ws_size,
                              hipStream_t stream) {
    const float* feat = (const float*)d_in[0];   // [8,64,256,512] f32
    const float* wgt  = (const float*)d_in[1];   // [1,64,3,3]     f32
    const int*   gi   = (const int*)d_in[2];     // [256,512,9]    i32
    const int*   gj   = (const int*)d_in[3];     // [256,512,9]    i32
    float*       out  = (float*)d_out;           // [8,1,512,1024] f32

    dim3 grid(Bn * Hn);   // 2048 blocks: one per (b, h) row
    dim3 block(256);      // 8 wave32
    hipLaunchKernelGGL(sphere_conv_wmma_kernel, grid, block, 0, stream,
                       feat, wgt, gi, gj, out);
}